// FrameSampler_24867860644429
// MI455X (gfx1250) — compile-verified
//
#include <hip/hip_runtime.h>
#include <math.h>

// ---------------------------------------------------------------------------
// FrameSampler pipeline for gfx1250 (MI455X).
//
// Roofline: act_maps (8x64x256x256 fp32 = 134MB) write-bound; everything else
// is tiny. The 'bpc,bfac->bfap' einsum (M=65536, N=192, K=3) is done with
// V_WMMA_F32_16X16X4_F32 (K padded to 4 with zeros) -> exact fp32 math.
// Top-k stages use iterative block argmax matching JAX top_k ordering
// (value desc, index asc on ties).
// ---------------------------------------------------------------------------

#define EPSF 1e-8f
#define NSEG 2048
#define NVSEL 128
#define NHSEL 256
#define NJ 512
#define NHL 256
#define NHP 512
#define NFR 64
#define HW 65536

// d_out float offsets (flat concat of reference tuple, 8 batches):
#define OFF_VSEGS   0        // 8*128*4  = 4096
#define OFF_HSEGS   4096     // 8*256*4  = 8192
#define OFF_JUNCS   12288    // 8*512*3  = 12288
#define OFF_HSEGSP  24576    // 8*256*4  = 8192
#define OFF_HSEGSM  32768    // 8*256*4  = 8192
#define OFF_HPTSP   40960    // 8*512*3  = 12288
#define OFF_HPTSM   53248    // 8*512*3  = 12288
#define OFF_FRAMES  65536    // 8*64*3*3 = 4608
#define OFF_FOCALS  70144    // 8*64     = 512
#define OFF_ACT     70656    // 8*64*65536

typedef __attribute__((ext_vector_type(2))) float v2f;
typedef __attribute__((ext_vector_type(8))) float v8f;

__device__ __forceinline__ float safef(float w) { return fabsf(w) < EPSF ? EPSF : w; }

// Block-wide argmax: value descending, index ascending tie-break (JAX top_k).
template <int NTH, class GET>
__device__ __forceinline__ int block_argmax(int n, GET get, float* redV, int* redI) {
  const int tid = threadIdx.x;
  float bv = -INFINITY;
  int bi = n;
  for (int i = tid; i < n; i += NTH) {
    float v = get(i);
    if (v > bv || (v == bv && i < bi)) { bv = v; bi = i; }
  }
  redV[tid] = bv;
  redI[tid] = bi;
  __syncthreads();
  for (int s = NTH >> 1; s > 0; s >>= 1) {
    if (tid < s) {
      float ov = redV[tid + s];
      int   oi = redI[tid + s];
      if (ov > redV[tid] || (ov == redV[tid] && oi < redI[tid])) {
        redV[tid] = ov; redI[tid] = oi;
      }
    }
    __syncthreads();
  }
  int sel = redI[0];
  __syncthreads();
  return sel;
}

// ---------------------------------------------------------------------------
// Kernel 1: init_filter_segs  (one block per batch)
// ---------------------------------------------------------------------------
__global__ __launch_bounds__(256) void k_init_filter(const float* __restrict__ segs,
                                                     const float* __restrict__ zvp,
                                                     float* __restrict__ out) {
  const int b = blockIdx.x, tid = threadIdx.x;
  __shared__ float sV[NSEG];
  __shared__ float sH[NSEG];
  __shared__ float redV[256];
  __shared__ int   redI[256];

  float zs  = safef(zvp[b * 3 + 2]);
  float z2x = zvp[b * 3 + 0] / zs;
  float z2y = zvp[b * 3 + 1] / zs;

  for (int i = tid; i < NSEG; i += 256) {
    const float* s = segs + ((size_t)b * NSEG + i) * 4;
    float x1 = s[0], y1 = s[1], x2 = s[2], y2 = s[3];
    float dx = x2 - x1, dy = y2 - y1;
    float len = sqrtf(dx * dx + dy * dy + EPSF);
    float dnx = dx / len, dny = dy / len;
    float mx = 0.5f * (x1 + x2), my = 0.5f * (y1 + y2);
    float tx = z2x - mx, ty = z2y - my;
    float tn = sqrtf(tx * tx + ty * ty + EPSF);
    float ca = fabsf(dnx * (tx / tn) + dny * (ty / tn));
    sV[i] = ca * len;
    sH[i] = (1.0f - ca) * len;
  }
  __syncthreads();

  for (int it = 0; it < NVSEL; ++it) {
    int sel = block_argmax<256>(NSEG, [&](int i) { return sV[i]; }, redV, redI);
    if (tid < 4) out[OFF_VSEGS + ((size_t)b * NVSEL + it) * 4 + tid] =
        segs[((size_t)b * NSEG + sel) * 4 + tid];
    if (tid == 0) sV[sel] = -INFINITY;
    __syncthreads();
  }
  for (int it = 0; it < NHSEL; ++it) {
    int sel = block_argmax<256>(NSEG, [&](int i) { return sH[i]; }, redV, redI);
    if (tid < 4) out[OFF_HSEGS + ((size_t)b * NHSEL + it) * 4 + tid] =
        segs[((size_t)b * NSEG + sel) * 4 + tid];
    if (tid == 0) sH[sel] = -INFINITY;
    __syncthreads();
  }
}

// ---------------------------------------------------------------------------
// Kernel 2: junc_filtering  (one block per batch)
// ---------------------------------------------------------------------------
__global__ __launch_bounds__(256) void k_junc(float* __restrict__ out,
                                              float* __restrict__ wsScore,
                                              int* __restrict__ wsIh) {
  const int b = blockIdx.x, tid = threadIdx.x;
  __shared__ float vl[NVSEL * 3];
  __shared__ float hl[NHSEL * 3];
  __shared__ float vmid[NVSEL * 2];
  __shared__ float hmid[NHSEL * 2];
  __shared__ unsigned msk[(NVSEL * NHSEL) / 32];  // 1024 words
  __shared__ float redV[256];
  __shared__ int   redI[256];

  for (int i = tid; i < NVSEL; i += 256) {
    const float* s = out + OFF_VSEGS + ((size_t)b * NVSEL + i) * 4;
    float x1 = s[0], y1 = s[1], x2 = s[2], y2 = s[3];
    float lx = y1 - y2, ly = x2 - x1, lz = x1 * y2 - y1 * x2;
    float n = sqrtf(lx * lx + ly * ly + EPSF);
    vl[i * 3 + 0] = lx / n; vl[i * 3 + 1] = ly / n; vl[i * 3 + 2] = lz / n;
    vmid[i * 2 + 0] = 0.5f * (x1 + x2);
    vmid[i * 2 + 1] = 0.5f * (y1 + y2);
  }
  for (int i = tid; i < NHSEL; i += 256) {
    const float* s = out + OFF_HSEGS + ((size_t)b * NHSEL + i) * 4;
    float x1 = s[0], y1 = s[1], x2 = s[2], y2 = s[3];
    float lx = y1 - y2, ly = x2 - x1, lz = x1 * y2 - y1 * x2;
    float n = sqrtf(lx * lx + ly * ly + EPSF);
    hl[i * 3 + 0] = lx / n; hl[i * 3 + 1] = ly / n; hl[i * 3 + 2] = lz / n;
    hmid[i * 2 + 0] = 0.5f * (x1 + x2);
    hmid[i * 2 + 1] = 0.5f * (y1 + y2);
  }
  for (int i = tid; i < (NVSEL * NHSEL) / 32; i += 256) msk[i] = 0u;
  __syncthreads();

  float* sc = wsScore + (size_t)b * (NVSEL * NHSEL);
  for (int idx = tid; idx < NVSEL * NHSEL; idx += 256) {
    int i = idx >> 8, j = idx & 255;
    float ax = vl[i * 3 + 0], ay = vl[i * 3 + 1], az = vl[i * 3 + 2];
    float bx = hl[j * 3 + 0], by = hl[j * 3 + 1], bz = hl[j * 3 + 2];
    float jx = ay * bz - az * by;
    float jy = az * bx - ax * bz;
    float jz = ax * by - ay * bx;
    float zs = safef(jz);
    float px = jx / zs, py = jy / zs;
    float dvx = px - vmid[i * 2 + 0], dvy = py - vmid[i * 2 + 1];
    float dhx = px - hmid[j * 2 + 0], dhy = py - hmid[j * 2 + 1];
    sc[idx] = -(sqrtf(dvx * dvx + dvy * dvy + EPSF) +
                sqrtf(dhx * dhx + dhy * dhy + EPSF));
  }
  __syncthreads();

  for (int it = 0; it < NJ; ++it) {
    int sel = block_argmax<256>(
        NVSEL * NHSEL,
        [&](int i) { return ((msk[i >> 5] >> (i & 31)) & 1u) ? -INFINITY : sc[i]; },
        redV, redI);
    if (tid == 0) {
      msk[sel >> 5] |= (1u << (sel & 31));
      int i = sel >> 8, j = sel & 255;
      float ax = vl[i * 3 + 0], ay = vl[i * 3 + 1], az = vl[i * 3 + 2];
      float bx = hl[j * 3 + 0], by = hl[j * 3 + 1], bz = hl[j * 3 + 2];
      float jx = ay * bz - az * by;
      float jy = az * bx - ax * bz;
      float jz = ax * by - ay * bx;
      float n = sqrtf(jx * jx + jy * jy + jz * jz + EPSF);
      float* J = out + OFF_JUNCS + ((size_t)b * NJ + it) * 3;
      J[0] = jx / n; J[1] = jy / n; J[2] = jz / n;
      wsIh[b * NJ + it] = j;
    }
    __syncthreads();
  }
}

// ---------------------------------------------------------------------------
// Kernel 3: filter_lines_by_dir  (one block per batch)
// ---------------------------------------------------------------------------
__global__ __launch_bounds__(256) void k_filter_dir(const float* __restrict__ zvp,
                                                    const int* __restrict__ wsIh,
                                                    float* __restrict__ out) {
  const int b = blockIdx.x, tid = threadIdx.x;
  __shared__ float jseg[NJ * 4];
  __shared__ float sP[NJ];
  __shared__ float sM[NJ];
  __shared__ float redV[256];
  __shared__ int   redI[256];

  float zs  = safef(zvp[b * 3 + 2]);
  float z2x = zvp[b * 3 + 0] / zs;
  float z2y = zvp[b * 3 + 1] / zs;

  for (int i = tid; i < NJ; i += 256) {
    int ih = wsIh[b * NJ + i];
    const float* s = out + OFF_HSEGS + ((size_t)b * NHSEL + ih) * 4;
    float x1 = s[0], y1 = s[1], x2 = s[2], y2 = s[3];
    jseg[i * 4 + 0] = x1; jseg[i * 4 + 1] = y1;
    jseg[i * 4 + 2] = x2; jseg[i * 4 + 3] = y2;
    float dx = x2 - x1, dy = y2 - y1;
    float len = sqrtf(dx * dx + dy * dy + EPSF);
    float mx = 0.5f * (x1 + x2), my = 0.5f * (y1 + y2);
    float vx = z2x - mx, vy = z2y - my;
    float sgn = dx * vy - dy * vx;
    float sg = (sgn > 0.f) ? 1.f : ((sgn < 0.f) ? -1.f : 0.f);
    sP[i] = sg * len;
    sM[i] = -sg * len;
  }
  __syncthreads();

  for (int it = 0; it < NHL; ++it) {
    int sel = block_argmax<256>(NJ, [&](int i) { return sP[i]; }, redV, redI);
    if (tid < 4) out[OFF_HSEGSP + ((size_t)b * NHL + it) * 4 + tid] = jseg[sel * 4 + tid];
    if (tid == 0) sP[sel] = -INFINITY;
    __syncthreads();
  }
  for (int it = 0; it < NHL; ++it) {
    int sel = block_argmax<256>(NJ, [&](int i) { return sM[i]; }, redV, redI);
    if (tid < 4) out[OFF_HSEGSM + ((size_t)b * NHL + it) * 4 + tid] = jseg[sel * 4 + tid];
    if (tid == 0) sM[sel] = -INFINITY;
    __syncthreads();
  }
}

// ---------------------------------------------------------------------------
// Kernel 4: generate_inter_pts  (grid = (batch, p/m))
// ---------------------------------------------------------------------------
__global__ __launch_bounds__(256) void k_interpts(float* __restrict__ out,
                                                  float* __restrict__ wsScore) {
  const int b = blockIdx.x, pm = blockIdx.y, tid = threadIdx.x;
  __shared__ float ll[NHL * 3];
  __shared__ unsigned msk[(NHL * NHL) / 32];  // 2048 words
  __shared__ float redV[256];
  __shared__ int   redI[256];

  const float* segIn = out + (pm == 0 ? OFF_HSEGSP : OFF_HSEGSM);
  float* ptsOut = out + (pm == 0 ? OFF_HPTSP : OFF_HPTSM);

  for (int i = tid; i < NHL; i += 256) {
    const float* s = segIn + ((size_t)b * NHL + i) * 4;
    float x1 = s[0], y1 = s[1], x2 = s[2], y2 = s[3];
    float lx = y1 - y2, ly = x2 - x1, lz = x1 * y2 - y1 * x2;
    float n = sqrtf(lx * lx + ly * ly + EPSF);
    ll[i * 3 + 0] = lx / n; ll[i * 3 + 1] = ly / n; ll[i * 3 + 2] = lz / n;
  }
  for (int i = tid; i < (NHL * NHL) / 32; i += 256) msk[i] = 0u;
  __syncthreads();

  float* sc = wsScore + ((size_t)(b * 2 + pm)) * (NHL * NHL);
  for (int idx = tid; idx < NHL * NHL; idx += 256) {
    int i = idx >> 8, j = idx & 255;
    float ax = ll[i * 3 + 0], ay = ll[i * 3 + 1], az = ll[i * 3 + 2];
    float bx = ll[j * 3 + 0], by = ll[j * 3 + 1], bz = ll[j * 3 + 2];
    float cx = ay * bz - az * by;
    float cy = az * bx - ax * bz;
    float cz = ax * by - ay * bx;
    float nrm = sqrtf(cx * cx + cy * cy + cz * cz + EPSF);
    sc[idx] = (i == j) ? -1000000000.0f : nrm;
  }
  __syncthreads();

  for (int it = 0; it < NHP; ++it) {
    int sel = block_argmax<256>(
        NHL * NHL,
        [&](int i) { return ((msk[i >> 5] >> (i & 31)) & 1u) ? -INFINITY : sc[i]; },
        redV, redI);
    if (tid == 0) {
      msk[sel >> 5] |= (1u << (sel & 31));
      int i = sel >> 8, j = sel & 255;
      float ax = ll[i * 3 + 0], ay = ll[i * 3 + 1], az = ll[i * 3 + 2];
      float bx = ll[j * 3 + 0], by = ll[j * 3 + 1], bz = ll[j * 3 + 2];
      float cx = ay * bz - az * by;
      float cy = az * bx - ax * bz;
      float cz = ax * by - ay * bx;
      float n = sqrtf(cx * cx + cy * cy + cz * cz + EPSF);
      float* P = ptsOut + ((size_t)b * NHP + it) * 3;
      P[0] = cx / n; P[1] = cy / n; P[2] = cz / n;
    }
    __syncthreads();
  }
}

// ---------------------------------------------------------------------------
// Kernel 5: generate_frames  (one block per batch, 64 threads = frames)
// ---------------------------------------------------------------------------
__global__ __launch_bounds__(64) void k_frames(const float* __restrict__ zvps,
                                               float* __restrict__ out) {
  const int b = blockIdx.x, k = threadIdx.x;
  const float* z = zvps + ((size_t)b * 32 + (k & 31)) * 3;
  int hidx = (k >> 5) & 511;  // 0 or 1 for k<64
  const float* hbase = ((k & 1) == 0) ? (out + OFF_HPTSP) : (out + OFF_HPTSM);
  const float* h = hbase + ((size_t)b * NHP + hidx) * 3;

  float zs = safef(z[2]);
  float z2x = z[0] / zs, z2y = z[1] / zs;
  float hs = safef(h[2]);
  float h2x = h[0] / hs, h2y = h[1] / hs;
  float dt = z2x * h2x + z2y * h2y;
  float f = sqrtf(fminf(fmaxf(-dt, 0.01f), 10000.0f));

  float dzn = sqrtf(z2x * z2x + z2y * z2y + f * f + EPSF);
  float dzx = z2x / dzn, dzy = z2y / dzn, dzz = f / dzn;
  float dhn = sqrtf(h2x * h2x + h2y * h2y + f * f + EPSF);
  float dhx = h2x / dhn, dhy = h2y / dhn, dhz = f / dhn;
  float pr = dhx * dzx + dhy * dzy + dhz * dzz;
  float txx = dhx - pr * dzx, txy = dhy - pr * dzy, txz = dhz - pr * dzz;
  float tn = sqrtf(txx * txx + txy * txy + txz * txz + EPSF);
  float dxx = txx / tn, dxy = txy / tn, dxz = txz / tn;
  float dyx = dzy * dxz - dzz * dxy;
  float dyy = dzz * dxx - dzx * dxz;
  float dyz = dzx * dxy - dzy * dxx;

  float* F = out + OFF_FRAMES + (size_t)(b * NFR + k) * 9;
  F[0] = dxx; F[1] = dxy; F[2] = dxz;
  F[3] = dyx; F[4] = dyy; F[5] = dyz;
  F[6] = dzx; F[7] = dzy; F[8] = dzz;
  out[OFF_FOCALS + b * NFR + k] = f;
}

// ---------------------------------------------------------------------------
// Kernel 6: generate_active_map with V_WMMA_F32_16X16X4_F32
//   grid = (64 pixel-chunks, 8 batches), block = 128 (4 waves).
//   Each wave: 16-pixel strip x 192 vp rows = 12 WMMAs (K=3 padded to 4).
// ---------------------------------------------------------------------------
#define RSTR 196  // padded LDS row stride (floats) for 192 vp rows

__global__ __launch_bounds__(128) void k_actmap(const float* __restrict__ seg_map,
                                                const float* __restrict__ seg_mask,
                                                float* __restrict__ out) {
  const int b = blockIdx.y;
  const int tid = threadIdx.x;
  const int lane = tid & 31;
  const int wv = tid >> 5;  // wave 0..3
  __shared__ float vpL[192 * 4];
  __shared__ float res[64 * RSTR];

  const float* frames = out + OFF_FRAMES;
  const float* focals = out + OFF_FOCALS;
  float* act = out + OFF_ACT;

  // Build padded vp (192 rows x 4 comps, comp3 = 0) in LDS.
  for (int r = tid; r < 192; r += 128) {
    int f = r / 3, a = r - 3 * f;
    const float* R = frames + ((size_t)(b * NFR + f) * 3 + a) * 3;
    float foc = focals[b * NFR + f];
    float vx = foc * R[0], vy = foc * R[1], vz = R[2];
    float n = sqrtf(vx * vx + vy * vy + vz * vz + EPSF);
    vpL[r * 4 + 0] = vx / n;
    vpL[r * 4 + 1] = vy / n;
    vpL[r * 4 + 2] = vz / n;
    vpL[r * 4 + 3] = 0.0f;
  }
  __syncthreads();

  // A-matrix fragments (16x4 f32): lane M = lane%16; K pair by lane half.
  const int hi = lane >> 4;
  const int lp = lane & 15;
  const int k0 = hi * 2;
  float a0[12], a1[12];
#pragma unroll
  for (int t = 0; t < 12; ++t) {
    int m = t * 16 + lp;
    a0[t] = vpL[m * 4 + k0];
    a1[t] = vpL[m * 4 + k0 + 1];
  }

  const int pixPerBlock = HW / gridDim.x;  // 1024 with 64 blocks
  const int base0 = blockIdx.x * pixPerBlock;

  for (int it = 0; it < pixPerBlock / 64; ++it) {
    const int chunk = base0 + it * 64;
    const int pg = chunk + wv * 16 + lp;  // this lane's pixel

    // B-matrix fragment (4x16 f32): lane N = lane%16; K pair by lane half.
    float lx = seg_map[((size_t)b * 3 + 0) * HW + pg];
    float ly = seg_map[((size_t)b * 3 + 1) * HW + pg];
    float lz = seg_map[((size_t)b * 3 + 2) * HW + pg];
    float n2 = sqrtf(lx * lx + ly * ly + EPSF);
    v2f bv;
    bv.x = hi ? (lz / n2) : (lx / n2);
    bv.y = hi ? 0.0f : (ly / n2);

    const int pl = wv * 16 + lp;  // chunk-local pixel
#pragma unroll
    for (int t = 0; t < 12; ++t) {
      v2f av;
      av.x = a0[t];
      av.y = a1[t];
      v8f cz = {0.f, 0.f, 0.f, 0.f, 0.f, 0.f, 0.f, 0.f};
      // D[m][n] = dot(vp_row[16t+m], line[pixel n]) -- exact fp32.
      v8f c = __builtin_amdgcn_wmma_f32_16x16x4_f32(
          /*neg_a=*/false, av, /*neg_b=*/false, bv,
          /*c_mod=*/(short)0, cz, /*reuse_a=*/false, /*reuse_b=*/false);
#pragma unroll
      for (int r = 0; r < 8; ++r) {
        // VGPR r holds row (r + 8*hi) of this 16-row tile, col = lane%16.
        res[pl * RSTR + t * 16 + r + 8 * hi] = c[r];
      }
    }
    __syncthreads();

    // act[b,f,p] = exp(-min_a cos^2 / (2 sigma^2)) * mask[p]
    const int p_local = tid & 63;
    const int pgo = chunk + p_local;
    const float mk = seg_mask[(size_t)b * HW + pgo];
    for (int f = tid >> 6; f < NFR; f += 2) {
      float q0 = res[p_local * RSTR + 3 * f + 0];
      float q1 = res[p_local * RSTR + 3 * f + 1];
      float q2 = res[p_local * RSTR + 3 * f + 2];
      q0 *= q0; q1 *= q1; q2 *= q2;
      float mn = fminf(q0, fminf(q1, q2));
      act[(size_t)(b * NFR + f) * HW + pgo] = expf(-mn / 0.02f) * mk;
    }
    __syncthreads();
  }
}

// ---------------------------------------------------------------------------
// Host launcher
// ---------------------------------------------------------------------------
extern "C" void kernel_launch(void* const* d_in, const int* in_sizes, int n_in,
                              void* d_out, int out_size, void* d_ws, size_t ws_size,
                              hipStream_t stream) {
  (void)in_sizes; (void)n_in; (void)out_size; (void)ws_size;
  const float* segs     = (const float*)d_in[0];  // (8,2048,4)
  const float* zvp      = (const float*)d_in[1];  // (8,3)
  const float* zvps     = (const float*)d_in[2];  // (8,32,3)
  const float* seg_map  = (const float*)d_in[3];  // (8,3,256,256)
  const float* seg_mask = (const float*)d_in[4];  // (8,1,256,256)
  float* out = (float*)d_out;

  // Workspace layout: [0, 4MB) score scratch (reused), then ih indices.
  float* wsScore = (float*)d_ws;
  int*   wsIh    = (int*)((char*)d_ws + (size_t)8 * 2 * 65536 * sizeof(float));

  k_init_filter<<<8, 256, 0, stream>>>(segs, zvp, out);
  k_junc<<<8, 256, 0, stream>>>(out, wsScore, wsIh);
  k_filter_dir<<<8, 256, 0, stream>>>(zvp, wsIh, out);
  k_interpts<<<dim3(8, 2), 256, 0, stream>>>(out, wsScore);
  k_frames<<<8, 64, 0, stream>>>(zvps, out);
  k_actmap<<<dim3(64, 8), 128, 0, stream>>>(seg_map, seg_mask, out);
}